// ME_26001732010613
// MI455X (gfx1250) — compile-verified
//
#include <hip/hip_runtime.h>
#include <hip/hip_bf16.h>

// ---------------- CDNA5 WMMA (fp32, 16x16x4) helpers ----------------
typedef float v2f __attribute__((ext_vector_type(2)));
typedef float v8f __attribute__((ext_vector_type(8)));

__device__ __forceinline__ v8f wmma4(v2f a, v2f b, v8f c) {
  // D = A(16x4) * B(4x16) + C(16x16), fp32, wave32
  return __builtin_amdgcn_wmma_f32_16x16x4_f32(
      false, a, false, b, (short)0, c, false, false);
}

__device__ __forceinline__ float wsum32(float v) {
#pragma unroll
  for (int m = 16; m > 0; m >>= 1) v += __shfl_xor(v, m, 32);
  return v;
}

#define ZV8 {0.f,0.f,0.f,0.f,0.f,0.f,0.f,0.f}

// =====================================================================
// conv1: 1x1, 512 -> 256, inputs = concat(boundary, box), bn+relu
// GEMM per batch: W(256x512) x X(512x4096).  Wave tile: 16 co x 64 pix.
// =====================================================================
__global__ void __launch_bounds__(256) k_conv1(
    const float* __restrict__ bnd, const float* __restrict__ box,
    const float* __restrict__ w1, const float* __restrict__ g,
    const float* __restrict__ be, float* __restrict__ out)
{
  const int wv = threadIdx.x >> 5, lane = threadIdx.x & 31;
  const int gw = blockIdx.x * 8 + wv;          // 8192 wave tiles
  const int b = gw >> 10;
  const int cot = (gw >> 6) & 15;
  const int pt  = gw & 63;
  const int co0 = cot << 4, p0 = pt << 6;
  const int m = lane & 15, half = lane >> 4, kk = half * 2;
  const int n = m;

  v8f acc[4] = {ZV8, ZV8, ZV8, ZV8};
  for (int ci0 = 0; ci0 < 512; ci0 += 4) {
    const float* src = (ci0 < 256) ? bnd : box;
    const int cc = ci0 & 255;
    v2f a = *(const v2f*)(w1 + (size_t)(co0 + m) * 512 + ci0 + kk);
    const float* xr = src + ((size_t)(b * 256 + cc + kk)) * 4096 + p0;
    // stage next K-step's activation lines into cache (global_prefetch_b8)
    if (ci0 + 4 < 512 && cc + kk + 4 < 256)
      __builtin_prefetch(xr + 4 * 4096 + n, 0, 3);
#pragma unroll
    for (int s = 0; s < 4; ++s) {
      v2f bb;
      bb.x = xr[s * 16 + n];
      bb.y = xr[4096 + s * 16 + n];
      acc[s] = wmma4(a, bb, acc[s]);
    }
  }
  const float rn = rsqrtf(1.00001f);
  const int mb = half * 8;
#pragma unroll
  for (int r = 0; r < 8; ++r) {
    const int co = co0 + mb + r;
    const float sc = g[co] * rn, bi = be[co];
#pragma unroll
    for (int s = 0; s < 4; ++s) {
      float v = fmaxf(acc[s][r] * sc + bi, 0.f);
      out[((size_t)(b * 256 + co)) * 4096 + p0 + s * 16 + n] = v;
    }
  }
}

// =====================================================================
// tap conv: kh x kw (dilated), 256 -> 256, bn+relu, optional +addsrc.
// Wave tile: 16 co x 64 w (one full image row).
// =====================================================================
__global__ void __launch_bounds__(256) k_conv_tap(
    const float* __restrict__ x, const float* __restrict__ wgt,
    const float* __restrict__ g, const float* __restrict__ be,
    const float* __restrict__ addsrc, float* __restrict__ out,
    int kh, int kw, int dil, int ph, int pw)
{
  const int wv = threadIdx.x >> 5, lane = threadIdx.x & 31;
  const int gw = blockIdx.x * 8 + wv;          // 8192 wave tiles
  const int b = gw >> 10;
  const int cot = (gw >> 6) & 15;
  const int h = gw & 63;
  const int co0 = cot << 4;
  const int m = lane & 15, half = lane >> 4, kk = half * 2;
  const int n = m;
  const int khw = kh * kw;

  v8f acc[4] = {ZV8, ZV8, ZV8, ZV8};
  const float* xb = x + ((size_t)b * 256) * 4096;

  for (int ky = 0; ky < kh; ++ky) {
    const int dy = ky * dil - ph;
    const int hh = h + dy;
    if (hh < 0 || hh >= 64) continue;          // wave-uniform
    for (int kx = 0; kx < kw; ++kx) {
      const int dx = kx * dil - pw;
      for (int ci0 = 0; ci0 < 256; ci0 += 4) {
        v2f a;
        a.x = wgt[((size_t)((co0 + m) * 256 + ci0 + kk)) * khw + ky * kw + kx];
        a.y = wgt[((size_t)((co0 + m) * 256 + ci0 + kk + 1)) * khw + ky * kw + kx];
        const float* xr = xb + ((size_t)(ci0 + kk)) * 4096 + hh * 64;
        if (ci0 + 4 < 256)
          __builtin_prefetch(xr + 4 * 4096 + n, 0, 3);   // global_prefetch_b8
#pragma unroll
        for (int s = 0; s < 4; ++s) {
          const int ww = s * 16 + n + dx;
          v2f bb;
          if (ww >= 0 && ww < 64) { bb.x = xr[ww]; bb.y = xr[4096 + ww]; }
          else                    { bb.x = 0.f;    bb.y = 0.f; }
          acc[s] = wmma4(a, bb, acc[s]);
        }
      }
    }
  }
  const float rn = rsqrtf(1.00001f);
  const int mb = half * 8;
#pragma unroll
  for (int r = 0; r < 8; ++r) {
    const int co = co0 + mb + r;
    const float sc = g[co] * rn, bi = be[co];
#pragma unroll
    for (int s = 0; s < 4; ++s) {
      const size_t oi = ((size_t)(b * 256 + co)) * 4096 + h * 64 + s * 16 + n;
      float v = fmaxf(acc[s][r] * sc + bi, 0.f);
      if (addsrc) v += addsrc[oi];
      out[oi] = v;
    }
  }
}

// =====================================================================
// projection GEMM: out(32768 x 256) = A(32768 x 256) * Wt^T, Wt(256x256)
// Wave tile: 16 tokens x 64 co.
// =====================================================================
__global__ void __launch_bounds__(256) k_proj(
    const float* __restrict__ A, const float* __restrict__ Wt,
    float* __restrict__ out)
{
  const int wv = threadIdx.x >> 5, lane = threadIdx.x & 31;
  const int gw = blockIdx.x * 8 + wv;          // 8192 wave tiles
  const int mt = gw >> 2, nt = gw & 3;
  const int t0 = mt << 4, co0 = nt << 6;
  const int m = lane & 15, half = lane >> 4, kk = half * 2;
  const int n = m;

  v8f acc[4] = {ZV8, ZV8, ZV8, ZV8};
  for (int k0 = 0; k0 < 256; k0 += 4) {
    const float* ap = A + (size_t)(t0 + m) * 256 + k0 + kk;
    if (k0 + 64 < 256)
      __builtin_prefetch(ap + 64, 0, 3);       // global_prefetch_b8 (next lines)
    v2f a = *(const v2f*)ap;
#pragma unroll
    for (int s = 0; s < 4; ++s) {
      v2f bb = *(const v2f*)(Wt + (size_t)(co0 + s * 16 + n) * 256 + k0 + kk);
      acc[s] = wmma4(a, bb, acc[s]);
    }
  }
  const int mb = half * 8;
#pragma unroll
  for (int r = 0; r < 8; ++r)
#pragma unroll
    for (int s = 0; s < 4; ++s)
      out[(size_t)(t0 + mb + r) * 256 + co0 + s * 16 + n] = acc[s][r];
}

// =====================================================================
// grouped 1x1 reduce-in: concat(bnd,box) 512 -> 256, groups=32
// =====================================================================
__global__ void __launch_bounds__(256) k_group_in(
    const float* __restrict__ bnd, const float* __restrict__ box,
    const float* __restrict__ wgt, const float* __restrict__ bias,
    float* __restrict__ out)
{
  __shared__ float ts[16][257];
  const int t = threadIdx.x;
  const int blk = blockIdx.x;       // 8 * 32 * 16
  const int b = blk >> 9;
  const int gg = (blk >> 4) & 31;
  const int p0 = (blk & 15) << 8;
#pragma unroll
  for (int rep = 0; rep < 16; ++rep) {
    const int idx = rep * 256 + t;
    const int i = idx >> 8, p = idx & 255;
    const int c = gg * 16 + i;
    const float* src = (c < 256) ? bnd : box;
    ts[i][p] = src[((size_t)(b * 256 + (c & 255))) * 4096 + p0 + p];
  }
  __syncthreads();
#pragma unroll
  for (int rep = 0; rep < 8; ++rep) {
    const int idx = rep * 256 + t;
    const int ol = idx >> 8, p = idx & 255;
    const int o = gg * 8 + ol;
    float acc = bias[o];
#pragma unroll
    for (int i = 0; i < 16; ++i) acc += wgt[o * 16 + i] * ts[i][p];
    out[((size_t)(b * 256 + o)) * 4096 + p0 + p] = acc;
  }
}

// =====================================================================
// grouped 1x1 reduce-out: concat(dense_em NCHW, hf (B,L,C)) -> d_out
// =====================================================================
__global__ void __launch_bounds__(256) k_group_out(
    const float* __restrict__ dem, const float* __restrict__ hf,
    const float* __restrict__ wgt, const float* __restrict__ bias,
    float* __restrict__ out)
{
  __shared__ float ts[16][257];
  const int t = threadIdx.x;
  const int blk = blockIdx.x;       // 8 * 32 * 16
  const int b = blk >> 9;
  const int gg = (blk >> 4) & 31;
  const int p0 = (blk & 15) << 8;
  if (gg < 16) {
#pragma unroll
    for (int rep = 0; rep < 16; ++rep) {
      const int idx = rep * 256 + t;
      const int i = idx >> 8, p = idx & 255;
      ts[i][p] = dem[((size_t)(b * 256 + gg * 16 + i)) * 4096 + p0 + p];
    }
  } else {
    const int ch0 = gg * 16 - 256;
#pragma unroll
    for (int rep = 0; rep < 16; ++rep) {
      const int idx = rep * 256 + t;
      const int i = idx & 15, p = idx >> 4;
      ts[i][p] = hf[((size_t)(b * 4096 + p0 + p)) * 256 + ch0 + i];
    }
  }
  __syncthreads();
#pragma unroll
  for (int rep = 0; rep < 8; ++rep) {
    const int idx = rep * 256 + t;
    const int ol = idx >> 8, p = idx & 255;
    const int o = gg * 8 + ol;
    float acc = bias[o];
#pragma unroll
    for (int i = 0; i < 16; ++i) acc += wgt[o * 16 + i] * ts[i][p];
    out[((size_t)(b * 256 + o)) * 4096 + p0 + p] = acc;
  }
}

// =====================================================================
// channel LayerNorm: hf NCHW (8,256,4096) -> hs (B, L, 256), eps 1e-5
// =====================================================================
__global__ void __launch_bounds__(256) k_chan_ln(
    const float* __restrict__ hf, const float* __restrict__ g,
    const float* __restrict__ bb, float* __restrict__ hs)
{
  __shared__ float ts[256][33];
  __shared__ float mu_s[32], rs_s[32];
  const int t = threadIdx.x;
  const int b = blockIdx.x >> 7;
  const int p0 = (blockIdx.x & 127) << 5;
  const float* src = hf + ((size_t)b * 256) * 4096 + p0;
#pragma unroll
  for (int rep = 0; rep < 32; ++rep) {
    const int idx = rep * 256 + t;
    const int c = idx >> 5, p = idx & 31;
    ts[c][p] = src[(size_t)c * 4096 + p];
  }
  __syncthreads();
  const int w = t >> 5, lane = t & 31;
  for (int pp = w; pp < 32; pp += 8) {
    float s = 0.f;
#pragma unroll
    for (int j = 0; j < 8; ++j) s += ts[lane + 32 * j][pp];
    const float mu = wsum32(s) * (1.f / 256.f);
    float v = 0.f;
#pragma unroll
    for (int j = 0; j < 8; ++j) { float d = ts[lane + 32 * j][pp] - mu; v += d * d; }
    v = wsum32(v) * (1.f / 256.f);
    if (lane == 0) { mu_s[pp] = mu; rs_s[pp] = rsqrtf(v + 1e-5f); }
  }
  __syncthreads();
  float* dst = hs + ((size_t)(b * 4096 + p0)) * 256;
#pragma unroll
  for (int rep = 0; rep < 32; ++rep) {
    const int idx = rep * 256 + t;
    const int p = idx >> 8, c = idx & 255;
    dst[(size_t)p * 256 + c] = (ts[c][p] - mu_s[p]) * rs_s[p] * g[c] + bb[c];
  }
}

// =====================================================================
// row LayerNorm: (32768, 256), eps 1e-6  (post-LN before wo GEMM)
// =====================================================================
__global__ void __launch_bounds__(256) k_row_ln(
    const float* __restrict__ in, const float* __restrict__ g,
    const float* __restrict__ bb, float* __restrict__ out)
{
  const int w = threadIdx.x >> 5, lane = threadIdx.x & 31;
  const size_t r = (size_t)blockIdx.x * 8 + w;
  const float* src = in + r * 256;
  float v[8], s = 0.f;
#pragma unroll
  for (int j = 0; j < 8; ++j) { v[j] = src[lane + 32 * j]; s += v[j]; }
  const float mu = wsum32(s) * (1.f / 256.f);
  float va = 0.f;
#pragma unroll
  for (int j = 0; j < 8; ++j) { float d = v[j] - mu; va += d * d; }
  va = wsum32(va) * (1.f / 256.f);
  const float rs = rsqrtf(va + 1e-6f);
  float* dst = out + r * 256;
#pragma unroll
  for (int j = 0; j < 8; ++j) {
    const int c = lane + 32 * j;
    dst[c] = (v[j] - mu) * rs * g[c] + bb[c];
  }
}

// ============== RoPE in-place on (B,L,NH*HD) buffer ==================
__global__ void k_rope(float* __restrict__ x)
{
  const int tid = blockIdx.x * 256 + threadIdx.x;  // 4,194,304 pairs
  const int hd = tid & 15;
  const int hh = (tid >> 4) & 7;
  const int l  = (tid >> 7) & 4095;
  const int b  = tid >> 19;
  const size_t idx = ((size_t)(b * 4096 + l)) * 256 + hh * 32 + hd;
  const float ang = (float)l * expf(-(float)hd * 0.5756462732485114f); // 10000^(-hd/16)
  const float c = cosf(ang), s = sinf(ang);
  const float x1 = x[idx], x2 = x[idx + 16];
  x[idx]      = x1 * c - x2 * s;
  x[idx + 16] = x2 * c + x1 * s;
}

// ============== ttt_lr = sigmoid(hs . lr_w[h] + lr_b[h]) =============
__global__ void k_lr(const float* __restrict__ hs, const float* __restrict__ lw,
                     const float* __restrict__ lb, float* __restrict__ lrs)
{
  const int tid = blockIdx.x * 256 + threadIdx.x;  // 262144
  const int l = tid & 4095;
  const int h = (tid >> 12) & 7;
  const int b = tid >> 15;
  const float* xr = hs + ((size_t)(b * 4096 + l)) * 256;
  const float* wr = lw + h * 256;
  float acc = 0.f;
  for (int c = 0; c < 256; ++c) acc += xr[c] * wr[c];
  acc += lb[h];
  lrs[((size_t)(b * 8 + h)) * 4096 + l] = 1.f / (1.f + expf(-acc));
}

// =====================================================================
// TTT scan: 64 blocks (one per (b,h) chain), 512 threads (wave = row).
// =====================================================================
__global__ void __launch_bounds__(512) k_scan(
    const float* __restrict__ XQ, const float* __restrict__ XK,
    const float* __restrict__ XV, const float* __restrict__ lrs_g,
    const float* __restrict__ W1, const float* __restrict__ b1,
    const float* __restrict__ lnw, const float* __restrict__ lnb,
    const float* __restrict__ tok_idx, float* __restrict__ out)
{
  __shared__ float Wst[32][33];
  __shared__ float bst[32];
  __shared__ float xq_s[16][33], xk_s[16][33], xv_s[16][33], gr_s[16][33];
  __shared__ float coef[16][16];
  __shared__ float lrs[16], tokv[16];

  const int t = threadIdx.x;
  const int w = t >> 5, lane = t & 31;
  const int b = blockIdx.x >> 3, h = blockIdx.x & 7;

  {
    const int i0 = t, i1 = t + 512;
    Wst[i0 >> 5][i0 & 31] = W1[h * 1024 + i0];
    Wst[i1 >> 5][i1 & 31] = W1[h * 1024 + i1];
    if (t < 32) bst[t] = b1[h * 32 + t];
    if (t < 16) {
      float tv = 1.f / (float)(t + 1) + tok_idx[t];
      tokv[t] = tv > 0.f ? tv : 0.f;
    }
  }
  const float gma = lnw[h * 32 + lane];
  const float bta = lnb[h * 32 + lane];
  __syncthreads();
  const float t15s = tokv[15] * (1.f / 32.f);

  const size_t base = ((size_t)b * 4096) * 256 + h * 32;
  const float* lrp = lrs_g + ((size_t)(b * 8 + h)) * 4096;

  for (int n = 0; n < 256; ++n) {
    const int l0 = n * 16;
    {
      const size_t src = base + (size_t)(l0 + w) * 256 + lane;
      xq_s[w][lane] = XQ[src];
      xk_s[w][lane] = XK[src];
      xv_s[w][lane] = XV[src];
      if (t < 16) lrs[t] = lrp[l0 + t];
      if (n + 1 < 256) {                       // prefetch next minibatch tiles
        const size_t nsrc = src + 16 * 256;
        __builtin_prefetch(XQ + nsrc, 0, 3);
        __builtin_prefetch(XK + nsrc, 0, 3);
        __builtin_prefetch(XV + nsrc, 0, 3);
      }
    }
    __syncthreads();

    // ---- Z1 = xk @ W + b ; grad = ln_fused_l2_bwd row-wise ----
    float z1 = bst[lane];
#pragma unroll
    for (int k = 0; k < 32; ++k) z1 += xk_s[w][k] * Wst[k][lane];
    const float mu = wsum32(z1) * (1.f / 32.f);
    const float dz = z1 - mu;
    const float var = wsum32(dz * dz) * (1.f / 32.f);
    const float stdv = sqrtf(var + 1e-6f);
    const float xh = dz / stdv;
    const float target = xv_s[w][lane] - xk_s[w][lane];
    const float gy = (gma * xh + bta - target) * gma;
    const float sgy = wsum32(gy);
    const float sgyxh = wsum32(gy * xh);
    const float grad = (32.f * gy - sgy - xh * sgyxh) / (32.f * stdv);
    gr_s[w][lane] = grad;
    __syncthreads();

    // ---- coef[i][j] = et[i][j]*(Attn[i][j]+1)  (tril-masked) ----
    if (t < 256) {
      const int i = t >> 4, j = t & 15;
      float at = 0.f;
#pragma unroll
      for (int d = 0; d < 32; ++d) at += xq_s[i][d] * xk_s[j][d];
      coef[i][j] = (j <= i) ? tokv[i] * (1.f / 32.f) * lrs[j] * (at + 1.f) : 0.f;
    }
    __syncthreads();

    // ---- Z1_bar = xq@W + b - coef@grad ; out = xq + LN(Z1_bar) ----
    float zb = bst[lane];
#pragma unroll
    for (int k = 0; k < 32; ++k) zb += xq_s[w][k] * Wst[k][lane];
#pragma unroll
    for (int j = 0; j < 16; ++j) zb -= coef[w][j] * gr_s[j][lane];
    const float mu2 = wsum32(zb) * (1.f / 32.f);
    const float d2 = zb - mu2;
    const float v2 = wsum32(d2 * d2) * (1.f / 32.f);
    out[base + (size_t)(l0 + w) * 256 + lane] =
        xq_s[w][lane] + d2 * rsqrtf(v2 + 1e-6f) * gma + bta;
    __syncthreads();  // all reads of old W/b complete

    // ---- W -= (last*xk)^T @ grad ; b -= sum(last*grad) ----
    {
      const int k1 = w, k2 = w + 16;
      float a1 = 0.f, a2 = 0.f, ab = 0.f;
#pragma unroll
      for (int i = 0; i < 16; ++i) {
        const float li = t15s * lrs[i];
        const float gi = gr_s[i][lane];
        a1 += li * xk_s[i][k1] * gi;
        a2 += li * xk_s[i][k2] * gi;
        ab += li * gi;
      }
      Wst[k1][lane] -= a1;
      Wst[k2][lane] -= a2;
      if (w == 0) bst[lane] -= ab;
    }
    __syncthreads();
  }
}

// ============== tail copy: sparse_embeddings_box -> d_out ============
__global__ void k_copy(const float* __restrict__ s, float* __restrict__ d, int nn)
{
  const int i = blockIdx.x * 256 + threadIdx.x;
  if (i < nn) d[i] = s[i];
}

// =====================================================================
extern "C" void kernel_launch(void* const* d_in, const int* in_sizes, int n_in,
                              void* d_out, int out_size, void* d_ws, size_t ws_size,
                              hipStream_t stream)
{
  const float* bnd   = (const float*)d_in[0];
  const float* box   = (const float*)d_in[1];
  const float* hfreq = (const float*)d_in[2];
  const float* sparse= (const float*)d_in[3];
  const float* w_ri  = (const float*)d_in[4];
  const float* b_ri  = (const float*)d_in[5];
  const float* w_ro  = (const float*)d_in[6];
  const float* b_ro  = (const float*)d_in[7];
  const float* w_c1  = (const float*)d_in[8];
  const float* g1    = (const float*)d_in[9];
  const float* be1   = (const float*)d_in[10];
  const float* w_c2  = (const float*)d_in[11];
  const float* g2    = (const float*)d_in[12];
  const float* be2   = (const float*)d_in[13];
  const float* w_c3  = (const float*)d_in[14];
  const float* g3    = (const float*)d_in[15];
  const float* be3   = (const float*)d_in[16];
  const float* w_c4  = (const float*)d_in[17];
  const float* g4    = (const float*)d_in[18];
  const float* be4   = (const float*)d_in[19];
  const float* ln_g  = (const float*)d_in[20];
  const float* ln_b  = (const float*)d_in[21];
  const float* wq    = (const float*)d_in[22];
  const float* wk    = (const float*)d_in[23];
  const float* wv    = (const float*)d_in[24];
  const float* wo    = (const float*)d_in[25];
  const float* W1    = (const float*)d_in[26];
  const float* b1    = (const float*)d_in[27];
  const float* tlnw  = (const float*)d_in[28];
  const float* tlnb  = (const float*)d_in[29];
  const float* lr_w  = (const float*)d_in[30];
  const float* lr_b  = (const float*)d_in[31];
  const float* tok   = (const float*)d_in[32];
  const float* plng  = (const float*)d_in[33];
  const float* plnb  = (const float*)d_in[34];
  float* outp = (float*)d_out;

  float* ws = (float*)d_ws;
  const size_t S = (size_t)8 * 256 * 64 * 64;   // 8,388,608 floats
  float* slot0 = ws;            // tmp -> hs -> attn_out
  float* slot1 = ws + S;        // x ping / XQ / normed
  float* slot2 = ws + 2 * S;    // x pong / dense_em
  float* slot3 = ws + 3 * S;    // XK / hf_out
  float* slot4 = ws + 4 * S;    // XV
  float* lrs   = ws + 5 * S;    // 262,144 floats

  // ---- conv chain ----
  k_group_in<<<4096, 256, 0, stream>>>(bnd, box, w_ri, b_ri, slot0);          // tmp
  k_conv1<<<1024, 256, 0, stream>>>(bnd, box, w_c1, g1, be1, slot1);          // x1
  k_conv_tap<<<1024, 256, 0, stream>>>(slot1, w_c2, g2, be2, nullptr, slot2,
                                       1, 3, 1, 0, 1);                        // x2
  k_conv_tap<<<1024, 256, 0, stream>>>(slot2, w_c3, g3, be3, nullptr, slot1,
                                       3, 1, 1, 1, 0);                        // x3
  k_conv_tap<<<1024, 256, 0, stream>>>(slot1, w_c4, g4, be4, slot0, slot2,
                                       3, 3, 3, 3, 3);                        // dense_em

  // ---- TTT path ----
  k_chan_ln<<<1024, 256, 0, stream>>>(hfreq, ln_g, ln_b, slot0);              // hs
  k_proj<<<1024, 256, 0, stream>>>(slot0, wq, slot1);                         // XQ
  k_proj<<<1024, 256, 0, stream>>>(slot0, wk, slot3);                         // XK
  k_proj<<<1024, 256, 0, stream>>>(slot0, wv, slot4);                         // XV
  k_rope<<<16384, 256, 0, stream>>>(slot1);
  k_rope<<<16384, 256, 0, stream>>>(slot3);
  k_lr<<<1024, 256, 0, stream>>>(slot0, lr_w, lr_b, lrs);
  k_scan<<<64, 512, 0, stream>>>(slot1, slot3, slot4, lrs, W1, b1,
                                 tlnw, tlnb, tok, slot0);                     // attn_out
  k_row_ln<<<4096, 256, 0, stream>>>(slot0, plng, plnb, slot1);               // normed
  k_proj<<<1024, 256, 0, stream>>>(slot1, wo, slot3);                         // hf_out

  // ---- final grouped conv + tail ----
  k_group_out<<<4096, 256, 0, stream>>>(slot2, slot3, w_ro, b_ro, outp);
  k_copy<<<16, 256, 0, stream>>>(sparse, outp + S, 4096);

  (void)in_sizes; (void)n_in; (void)out_size; (void)ws_size;
}